// PointTokenizer_23699629540067
// MI455X (gfx1250) — compile-verified
//
#include <hip/hip_runtime.h>
#include <hip/hip_bf16.h>
#include <math.h>

#define NPTS    16384
#define BATCH   8
#define NGRP    512
#define GS      32
#define DIN_PAD 16
#define BIGF    1.0e10f
#define TOK_ELEMS (BATCH * NGRP * 384)

typedef __attribute__((ext_vector_type(2))) float v2f;
typedef __attribute__((ext_vector_type(8))) float v8f;

// -------------------------------------------------------------------------
// Kernel 1: Farthest-point sampling. One workgroup per batch; the whole
// point cloud (192KB) + distance array (64KB) live in LDS (<=320KB/WGP).
// -------------------------------------------------------------------------
__global__ void __launch_bounds__(1024) fps_kernel(const float* __restrict__ points,
                                                   int* __restrict__ cidx,
                                                   float* __restrict__ centers_out) {
    extern __shared__ float sm[];
    float* px   = sm;
    float* py   = sm + NPTS;
    float* pz   = sm + 2 * NPTS;
    float* dist = sm + 3 * NPTS;
    __shared__ float rval[32];
    __shared__ int   ridx[32];
    __shared__ int   s_far;

    const int b   = blockIdx.x;
    const int tid = threadIdx.x;
    const float* P = points + (size_t)b * NPTS * 6;

    for (int i = tid; i < NPTS; i += 1024) {
        px[i]   = P[i * 6 + 0];
        py[i]   = P[i * 6 + 1];
        pz[i]   = P[i * 6 + 2];
        dist[i] = BIGF;
    }
    __syncthreads();

    int far = 0;
    for (int it = 0; it < NGRP; ++it) {
        const float cx = px[far], cy = py[far], cz = pz[far];
        if (tid == 0) {
            cidx[b * NGRP + it] = far;
            centers_out[((size_t)b * NGRP + it) * 3 + 0] = cx;
            centers_out[((size_t)b * NGRP + it) * 3 + 1] = cy;
            centers_out[((size_t)b * NGRP + it) * 3 + 2] = cz;
        }
        float bv = -1.0f;
        int   bi = 0;
        for (int i = tid; i < NPTS; i += 1024) {
            const float dx = px[i] - cx, dy = py[i] - cy, dz = pz[i] - cz;
            const float d  = dx * dx + dy * dy + dz * dz;
            const float nd = fminf(dist[i], d);
            dist[i] = nd;
            if (nd > bv) { bv = nd; bi = i; }
        }
        // wave32 argmax reduce
        for (int off = 16; off > 0; off >>= 1) {
            const float ov = __shfl_down(bv, off, 32);
            const int   oi = __shfl_down(bi, off, 32);
            if (ov > bv) { bv = ov; bi = oi; }
        }
        if ((tid & 31) == 0) { rval[tid >> 5] = bv; ridx[tid >> 5] = bi; }
        __syncthreads();
        if (tid == 0) {
            float v = rval[0]; int ix = ridx[0];
            for (int w = 1; w < 32; ++w)
                if (rval[w] > v) { v = rval[w]; ix = ridx[w]; }
            s_far = ix;
        }
        __syncthreads();
        far = s_far;
    }
}

// -------------------------------------------------------------------------
// Kernel 2: kNN (32 nearest) + gather; writes padded X tiles [grp][32][16]:
// cols 0..2 = rel xyz, 3..8 = raw 6-channel features, 9..15 = 0.
// -------------------------------------------------------------------------
__global__ void __launch_bounds__(256) knn_kernel(const float* __restrict__ points,
                                                  const int* __restrict__ cidx,
                                                  float* __restrict__ Xout) {
    extern __shared__ float dist[];   // NPTS floats (64KB)
    __shared__ float rval[8];
    __shared__ int   ridx[8];
    __shared__ int   s_j;

    const int grp = blockIdx.x;
    const int b   = grp >> 9;
    const int tid = threadIdx.x;
    const float* P = points + (size_t)b * NPTS * 6;

    const int   ci = cidx[grp];
    const float cx = P[ci * 6 + 0], cy = P[ci * 6 + 1], cz = P[ci * 6 + 2];

    for (int i = tid; i < NPTS; i += 256) {
        const float dx = P[i * 6 + 0] - cx;
        const float dy = P[i * 6 + 1] - cy;
        const float dz = P[i * 6 + 2] - cz;
        dist[i] = dx * dx + dy * dy + dz * dz;
    }
    __syncthreads();

    for (int s = 0; s < GS; ++s) {
        float bv = 2.0f * BIGF;
        int   bi = 0;
        for (int i = tid; i < NPTS; i += 256) {
            const float d = dist[i];
            if (d < bv) { bv = d; bi = i; }
        }
        for (int off = 16; off > 0; off >>= 1) {
            const float ov = __shfl_down(bv, off, 32);
            const int   oi = __shfl_down(bi, off, 32);
            if (ov < bv) { bv = ov; bi = oi; }
        }
        if ((tid & 31) == 0) { rval[tid >> 5] = bv; ridx[tid >> 5] = bi; }
        __syncthreads();
        if (tid == 0) {
            float v = rval[0]; int ix = ridx[0];
            for (int w = 1; w < 8; ++w)
                if (rval[w] < v) { v = rval[w]; ix = ridx[w]; }
            s_j = ix;
            dist[ix] = 2.0f * BIGF;   // exclude from later passes
        }
        __syncthreads();
        const int j = s_j;
        if (tid < DIN_PAD) {
            float val;
            if (tid < 3) {
                const float c = (tid == 0) ? cx : ((tid == 1) ? cy : cz);
                val = P[j * 6 + tid] - c;
            } else if (tid < 9) {
                val = P[j * 6 + (tid - 3)];
            } else {
                val = 0.0f;
            }
            Xout[((size_t)grp * GS + s) * DIN_PAD + tid] = val;
        }
    }
}

// -------------------------------------------------------------------------
// Kernel 2b: repack a weight matrix [krows x ncols] into WMMA B-fragment-
// major layout: wp[frag][lane][2], frag = kt*NT + nt. Rows >= krows are
// zero-padded (bakes the layer-1 K=9->16 padding into data, no exec masks).
// Lane l supplies k = kt*4 + 2*(l>>4) + e, n = nt*16 + (l&15).
// -------------------------------------------------------------------------
__global__ void __launch_bounds__(64) repack_kernel(const float* __restrict__ w,
                                                    float* __restrict__ wp,
                                                    int NT, int ncols, int krows) {
    const int frag = blockIdx.x;          // kt*NT + nt
    const int kt   = frag / NT;
    const int nt   = frag - kt * NT;
    const int t    = threadIdx.x;         // 0..63  (= lane*2 + e)
    const int l    = t >> 1;
    const int e    = t & 1;
    const int k    = kt * 4 + ((l >> 4) << 1) + e;
    const int n    = nt * 16 + (l & 15);
    wp[(size_t)frag * 64 + t] = (k < krows) ? w[(size_t)k * ncols + n] : 0.0f;
}

// -------------------------------------------------------------------------
// Kernel 3: WMMA MLP 16->64->128->256->384 (f32 tensor path) + max-pool.
// One wave per group of 32 points. A fragments staged through LDS (b64
// typed loads); B fragments are single coalesced b64 loads from the packed
// L2-resident weight images.
// -------------------------------------------------------------------------
__global__ void __launch_bounds__(32) mlp_kernel(
    const float* __restrict__ X,
    const v2f* __restrict__ w1p, const float* __restrict__ b1,
    const float* __restrict__ g1, const float* __restrict__ be1,
    const float* __restrict__ m1, const float* __restrict__ var1,
    const v2f* __restrict__ w2p, const float* __restrict__ b2,
    const float* __restrict__ g2, const float* __restrict__ be2,
    const float* __restrict__ m2, const float* __restrict__ var2,
    const v2f* __restrict__ w3p, const float* __restrict__ b3,
    const float* __restrict__ g3, const float* __restrict__ be3,
    const float* __restrict__ m3, const float* __restrict__ var3,
    const v2f* __restrict__ w4p, const float* __restrict__ b4,
    float* __restrict__ tokens) {

    __shared__ float sA[32 * 256];   // h1 (64 cols), later h3 (256 cols)
    __shared__ float sB[32 * 128];   // h2 (128 cols)

    const int grp  = blockIdx.x;
    const int lane = threadIdx.x;
    const int lrow = lane & 15;      // col (B/C/D) or row (A)
    const int lhk  = lane >> 4;      // K half-select / row-half select

    const v2f* Xv  = (const v2f*)(X + (size_t)grp * GS * DIN_PAD);
    const v2f* sAv = (const v2f*)sA;
    const v2f* sBv = (const v2f*)sB;

    // ----- layer 1: X(32x16) @ W1pad(16x64) -> h1 in sA -----
    for (int nt = 0; nt < 4; ++nt) {
        v8f acc0 = {0.f,0.f,0.f,0.f,0.f,0.f,0.f,0.f};
        v8f acc1 = {0.f,0.f,0.f,0.f,0.f,0.f,0.f,0.f};
        for (int kt = 0; kt < 4; ++kt) {
            const int k0 = kt * 4 + lhk * 2;
            v2f a0 = Xv[(lrow * DIN_PAD + k0) >> 1];
            v2f a1 = Xv[((16 + lrow) * DIN_PAD + k0) >> 1];
            v2f bf = w1p[(size_t)(kt * 4 + nt) * 32 + lane];
            acc0 = __builtin_amdgcn_wmma_f32_16x16x4_f32(false, a0, false, bf, (short)0, acc0, false, false);
            acc1 = __builtin_amdgcn_wmma_f32_16x16x4_f32(false, a1, false, bf, (short)0, acc1, false, false);
        }
        const int   n  = nt * 16 + lrow;
        const float sc = g1[n] * rsqrtf(var1[n] + 1e-5f);
        const float sh = (b1[n] - m1[n]) * sc + be1[n];
        #pragma unroll
        for (int e = 0; e < 8; ++e) {
            const int r = e + lhk * 8;
            sA[r * 64 + n]        = fmaxf(acc0[e] * sc + sh, 0.0f);
            sA[(16 + r) * 64 + n] = fmaxf(acc1[e] * sc + sh, 0.0f);
        }
    }
    __syncthreads();

    // ----- layer 2: h1(32x64) @ W2(64x128) -> h2 in sB -----
    for (int nt = 0; nt < 8; ++nt) {
        v8f acc0 = {0.f,0.f,0.f,0.f,0.f,0.f,0.f,0.f};
        v8f acc1 = {0.f,0.f,0.f,0.f,0.f,0.f,0.f,0.f};
        for (int kt = 0; kt < 16; ++kt) {
            const int k0 = kt * 4 + lhk * 2;
            v2f a0 = sAv[(lrow * 64 + k0) >> 1];
            v2f a1 = sAv[((16 + lrow) * 64 + k0) >> 1];
            v2f bf = w2p[(size_t)(kt * 8 + nt) * 32 + lane];
            acc0 = __builtin_amdgcn_wmma_f32_16x16x4_f32(false, a0, false, bf, (short)0, acc0, false, false);
            acc1 = __builtin_amdgcn_wmma_f32_16x16x4_f32(false, a1, false, bf, (short)0, acc1, false, false);
        }
        const int   n  = nt * 16 + lrow;
        const float sc = g2[n] * rsqrtf(var2[n] + 1e-5f);
        const float sh = (b2[n] - m2[n]) * sc + be2[n];
        #pragma unroll
        for (int e = 0; e < 8; ++e) {
            const int r = e + lhk * 8;
            sB[r * 128 + n]        = fmaxf(acc0[e] * sc + sh, 0.0f);
            sB[(16 + r) * 128 + n] = fmaxf(acc1[e] * sc + sh, 0.0f);
        }
    }
    __syncthreads();

    // ----- layer 3: h2(32x128) @ W3(128x256) -> h3 in sA -----
    for (int nt = 0; nt < 16; ++nt) {
        v8f acc0 = {0.f,0.f,0.f,0.f,0.f,0.f,0.f,0.f};
        v8f acc1 = {0.f,0.f,0.f,0.f,0.f,0.f,0.f,0.f};
        for (int kt = 0; kt < 32; ++kt) {
            const int k0 = kt * 4 + lhk * 2;
            v2f a0 = sBv[(lrow * 128 + k0) >> 1];
            v2f a1 = sBv[((16 + lrow) * 128 + k0) >> 1];
            v2f bf = w3p[(size_t)(kt * 16 + nt) * 32 + lane];
            acc0 = __builtin_amdgcn_wmma_f32_16x16x4_f32(false, a0, false, bf, (short)0, acc0, false, false);
            acc1 = __builtin_amdgcn_wmma_f32_16x16x4_f32(false, a1, false, bf, (short)0, acc1, false, false);
        }
        const int   n  = nt * 16 + lrow;
        const float sc = g3[n] * rsqrtf(var3[n] + 1e-5f);
        const float sh = (b3[n] - m3[n]) * sc + be3[n];
        #pragma unroll
        for (int e = 0; e < 8; ++e) {
            const int r = e + lhk * 8;
            sA[r * 256 + n]        = fmaxf(acc0[e] * sc + sh, 0.0f);
            sA[(16 + r) * 256 + n] = fmaxf(acc1[e] * sc + sh, 0.0f);
        }
    }
    __syncthreads();

    // ----- layer 4: h3(32x256) @ W4(256x384) + bias, max over 32 rows -----
    for (int nt = 0; nt < 24; ++nt) {
        v8f acc0 = {0.f,0.f,0.f,0.f,0.f,0.f,0.f,0.f};
        v8f acc1 = {0.f,0.f,0.f,0.f,0.f,0.f,0.f,0.f};
        for (int kt = 0; kt < 64; ++kt) {
            const int k0 = kt * 4 + lhk * 2;
            v2f a0 = sAv[(lrow * 256 + k0) >> 1];
            v2f a1 = sAv[((16 + lrow) * 256 + k0) >> 1];
            v2f bf = w4p[(size_t)(kt * 24 + nt) * 32 + lane];
            acc0 = __builtin_amdgcn_wmma_f32_16x16x4_f32(false, a0, false, bf, (short)0, acc0, false, false);
            acc1 = __builtin_amdgcn_wmma_f32_16x16x4_f32(false, a1, false, bf, (short)0, acc1, false, false);
        }
        const int n = nt * 16 + lrow;
        float mx = -3.4e38f;
        #pragma unroll
        for (int e = 0; e < 8; ++e) mx = fmaxf(mx, fmaxf(acc0[e], acc1[e]));
        mx += b4[n];
        mx = fmaxf(mx, __shfl_xor(mx, 16, 32));   // combine row halves (same col)
        if (lane < 16) tokens[(size_t)grp * 384 + n] = mx;
    }
}

// -------------------------------------------------------------------------
// Kernel 4: positional MLP 3 -> 128 (exact gelu) -> 384, added into tokens.
// -------------------------------------------------------------------------
__global__ void __launch_bounds__(128) pos_kernel(const float* __restrict__ pw1,
                                                  const float* __restrict__ pb1,
                                                  const float* __restrict__ pw2,
                                                  const float* __restrict__ pb2,
                                                  const float* __restrict__ centers,
                                                  float* __restrict__ tokens) {
    __shared__ float p1[128];
    const int grp = blockIdx.x;
    const int t   = threadIdx.x;
    const float cx = centers[(size_t)grp * 3 + 0];
    const float cy = centers[(size_t)grp * 3 + 1];
    const float cz = centers[(size_t)grp * 3 + 2];
    const float s  = cx * pw1[t] + cy * pw1[128 + t] + cz * pw1[256 + t] + pb1[t];
    p1[t] = 0.5f * s * (1.0f + erff(s * 0.70710678118654752f));
    __syncthreads();
    for (int c = t; c < 384; c += 128) {
        float acc = pb2[c];
        for (int k = 0; k < 128; ++k) acc += p1[k] * pw2[k * 384 + c];
        tokens[(size_t)grp * 384 + c] += acc;
    }
}

// -------------------------------------------------------------------------
extern "C" void kernel_launch(void* const* d_in, const int* in_sizes, int n_in,
                              void* d_out, int out_size, void* d_ws, size_t ws_size,
                              hipStream_t stream) {
    const float* points = (const float*)d_in[0];
    const float* w1   = (const float*)d_in[1];
    const float* b1   = (const float*)d_in[2];
    const float* g1   = (const float*)d_in[3];
    const float* be1  = (const float*)d_in[4];
    const float* m1   = (const float*)d_in[5];
    const float* var1 = (const float*)d_in[6];
    const float* w2   = (const float*)d_in[7];
    const float* b2   = (const float*)d_in[8];
    const float* g2   = (const float*)d_in[9];
    const float* be2  = (const float*)d_in[10];
    const float* m2   = (const float*)d_in[11];
    const float* var2 = (const float*)d_in[12];
    const float* w3   = (const float*)d_in[13];
    const float* b3   = (const float*)d_in[14];
    const float* g3   = (const float*)d_in[15];
    const float* be3  = (const float*)d_in[16];
    const float* m3   = (const float*)d_in[17];
    const float* var3 = (const float*)d_in[18];
    const float* w4   = (const float*)d_in[19];
    const float* b4   = (const float*)d_in[20];
    const float* pw1  = (const float*)d_in[21];
    const float* pb1  = (const float*)d_in[22];
    const float* pw2  = (const float*)d_in[23];
    const float* pb2  = (const float*)d_in[24];

    float* tokens  = (float*)d_out;
    float* centers = tokens + (size_t)TOK_ELEMS;

    // workspace layout (bytes)
    char* ws   = (char*)d_ws;
    int*  cidx = (int*)ws;                                   // 16 KB
    float* X   = (float*)(ws + 16384);                       // 8 MB
    float* w1p = (float*)(ws + 16384 + 8388608);             // 16 KB  (4x4   frags)
    float* w2p = (float*)(ws + 16384 + 8388608 + 16384);     // 32 KB  (16x8  frags)
    float* w3p = (float*)(ws + 16384 + 8388608 + 49152);     // 128 KB (32x16 frags)
    float* w4p = (float*)(ws + 16384 + 8388608 + 180224);    // 384 KB (64x24 frags)

    // 1) FPS: one workgroup per batch; 256KB dynamic LDS (points + distances)
    fps_kernel<<<BATCH, 1024, 4 * NPTS * sizeof(float), stream>>>(points, cidx, centers);

    // 2) kNN + gather: one workgroup per center; 64KB dynamic LDS
    knn_kernel<<<BATCH * NGRP, 256, NPTS * sizeof(float), stream>>>(points, cidx, X);

    // 2b) repack weights into B-fragment-major images (zero-pads W1 K 9->16)
    repack_kernel<<< 4 * 4,   64, 0, stream>>>(w1, w1p,  4,  64,   9);
    repack_kernel<<<16 * 8,   64, 0, stream>>>(w2, w2p,  8, 128,  64);
    repack_kernel<<<32 * 16,  64, 0, stream>>>(w3, w3p, 16, 256, 128);
    repack_kernel<<<64 * 24,  64, 0, stream>>>(w4, w4p, 24, 384, 256);

    // 3) WMMA f32 MLP + maxpool: one wave per group
    mlp_kernel<<<BATCH * NGRP, 32, 0, stream>>>(X,
        (const v2f*)w1p, b1, g1, be1, m1, var1,
        (const v2f*)w2p, b2, g2, be2, m2, var2,
        (const v2f*)w3p, b3, g3, be3, m3, var3,
        (const v2f*)w4p, b4, tokens);

    // 4) positional encoding add
    pos_kernel<<<BATCH * NGRP, 128, 0, stream>>>(pw1, pb1, pw2, pb2, centers, tokens);
}